// Model_38508676776606
// MI455X (gfx1250) — compile-verified
//
#include <hip/hip_runtime.h>
#include <hip/hip_bf16.h>
#include <math.h>

// Problem constants (match reference setup_inputs)
#define Bn   8
#define Nn   4096
#define Dn   64
#define Sn   2048      // GROUPS
#define Kn   32        // KNEIGHBORS
#define CHn  67        // D + 3
#define OCn  131       // 2D + 3
#define NEWXYZ_ELEMS (Bn * Sn * 3)   // first output (new_xyz) flattened

typedef float v2f __attribute__((ext_vector_type(2)));
typedef float v8f __attribute__((ext_vector_type(8)));

// ---------- wave32 butterfly arg-reductions (tie-break: lowest index, matches jnp) ----------
__device__ __forceinline__ void argmax_reduce(float& v, int& i) {
#pragma unroll
  for (int off = 16; off > 0; off >>= 1) {
    float ov = __shfl_xor(v, off);
    int   oi = __shfl_xor(i, off);
    if (ov > v || (ov == v && oi < i)) { v = ov; i = oi; }
  }
}
__device__ __forceinline__ void argmin_reduce(float& v, int& i) {
#pragma unroll
  for (int off = 16; off > 0; off >>= 1) {
    float ov = __shfl_xor(v, off);
    int   oi = __shfl_xor(i, off);
    if (ov < v || (ov == v && oi < i)) { v = ov; i = oi; }
  }
}

// Async global->LDS copy of one dword per lane (gfx1250 GLOBAL_LOAD_ASYNC_TO_LDS_B32,
// tracked with ASYNCcnt). ldsoff = wave-relative LDS byte offset (low 32 bits of the
// generic LDS address per the flat-aperture rules); gaddr = global VA.
__device__ __forceinline__ void async_g2l_b32(unsigned ldsoff, const void* gaddr, int ioff) {
  unsigned long long a = (unsigned long long)gaddr;
  if (ioff == 0)
    asm volatile("global_load_async_to_lds_b32 %0, %1, off" :: "v"(ldsoff), "v"(a) : "memory");
  else if (ioff == 4)
    asm volatile("global_load_async_to_lds_b32 %0, %1, off offset:4" :: "v"(ldsoff), "v"(a) : "memory");
  else
    asm volatile("global_load_async_to_lds_b32 %0, %1, off offset:8" :: "v"(ldsoff), "v"(a) : "memory");
}
__device__ __forceinline__ void wait_async0() {
  asm volatile("s_wait_asynccnt 0" ::: "memory");
}

// =====================================================================================
// Kernel 1: farthest point sampling. One 1024-thread block per batch.
// xyz is staged AoS->SoA into LDS via the async Global->LDS engine (ASYNCcnt path),
// then 2047 sequential rounds of min-update + two-level wave32 argmax reduction.
// Writes fps_idx[B,S] and new_xyz (start of d_out).
// =====================================================================================
__global__ void fps_kernel(const float* __restrict__ xyz,
                           float* __restrict__ out,
                           int* __restrict__ fps_idx) {
  extern __shared__ float smem[];
  float* xs   = smem;
  float* ys   = xs + Nn;
  float* zs   = ys + Nn;
  float* dist = zs + Nn;
  float* rv   = dist + Nn;          // [32] per-wave best value
  int*   ri   = (int*)(rv + 32);    // [32] per-wave best index
  int*   bcast = ri + 32;           // [1]

  const int tid = threadIdx.x;
  const int b   = blockIdx.x;
  const float* xb = xyz + (size_t)b * Nn * 3;

  // Async-stage xyz (AoS in memory) to SoA x/y/z arrays in LDS; init dist with VALU/DS.
  for (int i = tid; i < Nn; i += blockDim.x) {
    const float* g = xb + (size_t)i * 3;
    async_g2l_b32((unsigned)(uintptr_t)&xs[i], g, 0);
    async_g2l_b32((unsigned)(uintptr_t)&ys[i], g, 4);
    async_g2l_b32((unsigned)(uintptr_t)&zs[i], g, 8);
    dist[i] = 1e10f;
  }
  wait_async0();
  __syncthreads();

  if (tid == 0) {
    fps_idx[b * Sn] = 0;
    out[(size_t)(b * Sn) * 3 + 0] = xs[0];
    out[(size_t)(b * Sn) * 3 + 1] = ys[0];
    out[(size_t)(b * Sn) * 3 + 2] = zs[0];
  }

  const int wid = tid >> 5, lane = tid & 31;
  int cur = 0;
  for (int s = 1; s < Sn; ++s) {
    const float cx = xs[cur], cy = ys[cur], cz = zs[cur];
    float bv = -1.0f;
    int   bi = 0x7fffffff;
    for (int i = tid; i < Nn; i += blockDim.x) {
      const float dx = xs[i] - cx, dy = ys[i] - cy, dz = zs[i] - cz;
      const float d  = dx * dx + dy * dy + dz * dz;
      const float dd = fminf(dist[i], d);
      dist[i] = dd;
      if (dd > bv || (dd == bv && i < bi)) { bv = dd; bi = i; }
    }
    argmax_reduce(bv, bi);
    if (lane == 0) { rv[wid] = bv; ri[wid] = bi; }
    __syncthreads();
    if (wid == 0) {
      float v = rv[lane];
      int   i = ri[lane];
      argmax_reduce(v, i);
      if (lane == 0) {
        bcast[0] = i;
        fps_idx[b * Sn + s] = i;
        out[(size_t)(b * Sn + s) * 3 + 0] = xs[i];
        out[(size_t)(b * Sn + s) * 3 + 1] = ys[i];
        out[(size_t)(b * Sn + s) * 3 + 2] = zs[i];
      }
    }
    __syncthreads();
    cur = bcast[0];
  }
}

// =====================================================================================
// Kernel 2: kNN. One 256-thread (8-wave) block per 16 anchors.
// sqd[16][4096] panel in LDS (256KB of the 320KB/WGP).
// Distances via V_WMMA_F32_16X16X4_F32:
//   A (16x4)  = [-2ax, -2ay, -2az, 1]        (per-anchor row)
//   B (4x16)  = [px, py, pz, |p|^2]^T        (per-point column)
//   C (16x16) = |a|^2 broadcast per row
//   D         = |a|^2 - 2 a.p + |p|^2 = squared distance tile
// B operands are built branch-free (all lanes load the xyz triple, v_cndmask selects
// the K-slice) so the WMMA loop runs with full EXEC and no saveexec juggling.
// Then 32 masked wave-argmin passes per anchor -> ascending-distance indices
// (same order + tie-break as jax.lax.top_k(-sqd, K)).
// =====================================================================================
__global__ void knn_kernel(const float* __restrict__ xyz,
                           const float* __restrict__ newxyz,  // = d_out head
                           int* __restrict__ knn_idx) {
  extern __shared__ float smem[];
  float* sqd = smem;                 // 16 * 4096
  float* ax  = sqd + 16 * Nn;        // 16
  float* ay  = ax + 16;
  float* az  = ay + 16;
  float* an  = az + 16;

  const int tid = threadIdx.x;
  const int b   = blockIdx.x >> 7;          // 128 groups of 16 anchors per batch
  const int a0  = (blockIdx.x & 127) * 16;

  if (tid < 16) {
    const float* p = newxyz + (size_t)(b * Sn + a0 + tid) * 3;
    const float x = p[0], y = p[1], z = p[2];
    ax[tid] = x; ay[tid] = y; az[tid] = z;
    an[tid] = x * x + y * y + z * z;
  }
  __syncthreads();

  const int  wid  = tid >> 5;
  const int  lane = tid & 31;
  const bool hiH  = (lane >= 16);
  const int  l15  = lane & 15;
  const int  mofs = hiH ? 8 : 0;

  // A matrix (16x4 f32): lanes 0-15 -> K=0 (vgpr0), K=1 (vgpr1); lanes 16-31 -> K=2, K=3.
  const float a_x = hiH ? (-2.0f * az[l15]) : (-2.0f * ax[l15]);
  const float a_y = hiH ? 1.0f              : (-2.0f * ay[l15]);
  v2f A; A.x = a_x; A.y = a_y;

  // C matrix (16x16 f32): vgpr r holds row M=r (lanes 0-15) / M=r+8 (lanes 16-31).
  v8f C;
#pragma unroll
  for (int r = 0; r < 8; ++r) C[r] = an[r + mofs];

  const float* xb = xyz + (size_t)b * Nn * 3;
  for (int t = wid; t < Nn / 16; t += 8) {
    const int p = t * 16 + l15;
    // Branch-free operand build: every lane loads its point's xyz triple (-> b96 load),
    // then selects its K-slice. Prefetch this wave's next tile (xyz is L2-resident).
    const float x = xb[p * 3 + 0];
    const float y = xb[p * 3 + 1];
    const float z = xb[p * 3 + 2];
    if (t + 8 < Nn / 16) __builtin_prefetch(&xb[(p + 128) * 3], 0, 0);
    const float nrm = x * x + y * y + z * z;
    v2f Bm;
    Bm.x = hiH ? z   : x;                   // rows K=2 / K=0
    Bm.y = hiH ? nrm : y;                   // rows K=3 / K=1
    v8f Dv = __builtin_amdgcn_wmma_f32_16x16x4_f32(
        /*neg_a=*/false, A, /*neg_b=*/false, Bm,
        /*c_mod=*/(short)0, C, /*reuse_a=*/false, /*reuse_b=*/false);
#pragma unroll
    for (int r = 0; r < 8; ++r)
      sqd[(r + mofs) * Nn + t * 16 + l15] = Dv[r];
  }
  __syncthreads();

  // Selection: each wave owns 2 anchors; 32 argmin+mask passes each.
  for (int m = 2 * wid; m < 2 * wid + 2; ++m) {
    volatile float* row = sqd + m * Nn;
    int* orow = knn_idx + (size_t)(b * Sn + a0 + m) * Kn;
    for (int j = 0; j < Kn; ++j) {
      float bv = 3.0e38f;
      int   bi = 0x7fffffff;
      for (int i = lane; i < Nn; i += 32) {
        const float v = row[i];
        if (v < bv) { bv = v; bi = i; }     // strict '<' keeps lowest index per lane
      }
      argmin_reduce(bv, bi);                // butterfly: all lanes hold the winner
      if (lane == 0) orow[j] = bi;
      if (lane == (bi & 31)) row[bi] = 3.4e38f;   // mask winner out
      asm volatile("s_wait_dscnt 0" ::: "memory"); // order LDS store before next pass
      __builtin_amdgcn_wave_barrier();
    }
  }
}

// =====================================================================================
// Kernel 3: per-anchor mean over K, plus per-anchor sum of squares of centered values.
// One 128-thread block per (b,s). Deterministic LDS tree reduce (no atomics).
// Note: sum_k (g - mean) == 0 exactly in fp32 (mean = sum * 2^-5 is exact), so the
// batch std only needs the sum of squares.
// =====================================================================================
__global__ void stats_kernel(const float* __restrict__ xyz,
                             const float* __restrict__ points,
                             const int* __restrict__ knn_idx,
                             float* __restrict__ meanbuf,
                             float* __restrict__ persumsq) {
  __shared__ int   sidx[Kn];
  __shared__ float red[128];
  const int tid = threadIdx.x;
  const int bs  = blockIdx.x;           // b*Sn + s
  const int b   = bs >> 11;

  if (tid < Kn) sidx[tid] = knn_idx[(size_t)bs * Kn + tid];
  __syncthreads();

  float sq = 0.0f;
  if (tid < CHn) {
    const int c = tid;
    float gk[Kn];
    float sum = 0.0f;
#pragma unroll
    for (int k = 0; k < Kn; ++k) {
      const int id = sidx[k];
      const float g = (c < Dn) ? points[((size_t)b * Nn + id) * Dn + c]
                               : xyz[((size_t)b * Nn + id) * 3 + (c - Dn)];
      gk[k] = g;
      sum += g;
    }
    const float mean = sum * (1.0f / (float)Kn);
    meanbuf[(size_t)bs * CHn + c] = mean;
#pragma unroll
    for (int k = 0; k < Kn; ++k) {
      const float d = gk[k] - mean;
      sq += d * d;
    }
  }
  red[tid] = sq;
  __syncthreads();
  for (int off = 64; off > 0; off >>= 1) {
    if (tid < off) red[tid] += red[tid + off];
    __syncthreads();
  }
  if (tid == 0) persumsq[bs] = red[0];
}

// =====================================================================================
// Kernel 3b: per-batch std = sqrt( sum(persumsq) / (S*K*CH - 1) ), deterministic tree.
// =====================================================================================
__global__ void std_reduce_kernel(const float* __restrict__ persumsq,
                                  float* __restrict__ stdbuf) {
  __shared__ float red[1024];
  const int tid = threadIdx.x;
  const int b   = blockIdx.x;
  red[tid] = persumsq[b * Sn + tid] + persumsq[b * Sn + tid + 1024];
  __syncthreads();
  for (int off = 512; off > 0; off >>= 1) {
    if (tid < off) red[tid] += red[tid + off];
    __syncthreads();
  }
  if (tid == 0) {
    const float n = (float)Sn * (float)Kn * (float)CHn;   // 4390912, exact in fp32
    stdbuf[b] = sqrtf(red[0] / (n - 1.0f));
  }
}

// =====================================================================================
// Kernel 4: write new_feat [B,S,K,131]. One 256-thread block per (b,s); threads < 131
// each own one output channel, looping over k. Channels 0..66 are the normalized
// grouped features; 67..130 are the broadcast anchor features.
// =====================================================================================
__global__ void output_kernel(const float* __restrict__ xyz,
                              const float* __restrict__ points,
                              const float* __restrict__ alpha,
                              const float* __restrict__ beta,
                              const int* __restrict__ fps_idx,
                              const int* __restrict__ knn_idx,
                              const float* __restrict__ meanbuf,
                              const float* __restrict__ stdbuf,
                              float* __restrict__ out) {
  __shared__ int   sidx[Kn];
  __shared__ float afeat[Dn];
  __shared__ float smean[CHn], salpha[CHn], sbeta[CHn];
  const int tid = threadIdx.x;
  const int bs  = blockIdx.x;
  const int b   = bs >> 11;

  if (tid < Kn) sidx[tid] = knn_idx[(size_t)bs * Kn + tid];
  const int fidx = fps_idx[bs];
  if (tid < Dn) afeat[tid] = points[((size_t)b * Nn + fidx) * Dn + tid];
  if (tid >= 128 && tid < 128 + CHn) {
    const int c = tid - 128;
    smean[c]  = meanbuf[(size_t)bs * CHn + c];
    salpha[c] = alpha[c];
    sbeta[c]  = beta[c];
  }
  const float inv = 1.0f / (stdbuf[b] + 1e-5f);
  __syncthreads();

  float* ob = out + NEWXYZ_ELEMS + (size_t)bs * Kn * OCn;
  if (tid < OCn) {
    for (int k = 0; k < Kn; ++k) {
      float v;
      if (tid < CHn) {
        const int id = sidx[k];
        const float g = (tid < Dn) ? points[((size_t)b * Nn + id) * Dn + tid]
                                   : xyz[((size_t)b * Nn + id) * 3 + (tid - Dn)];
        v = salpha[tid] * (g - smean[tid]) * inv + sbeta[tid];
      } else {
        v = afeat[tid - CHn];
      }
      ob[(size_t)k * OCn + tid] = v;
    }
  }
}

// =====================================================================================
// Host entry
// =====================================================================================
extern "C" void kernel_launch(void* const* d_in, const int* in_sizes, int n_in,
                              void* d_out, int out_size, void* d_ws, size_t ws_size,
                              hipStream_t stream) {
  const float* xyz    = (const float*)d_in[0];   // [8,4096,3]
  const float* points = (const float*)d_in[1];   // [8,4096,64]
  const float* alpha  = (const float*)d_in[2];   // [67]
  const float* beta   = (const float*)d_in[3];   // [67]
  float* out = (float*)d_out;                    // new_xyz (49152) ++ new_feat

  // Workspace layout (~6.6 MB total), fully (re)written every call.
  int*   fps_idx  = (int*)d_ws;                              // B*S
  int*   knn_idx  = fps_idx + Bn * Sn;                       // B*S*K
  float* meanbuf  = (float*)(knn_idx + Bn * Sn * Kn);        // B*S*CH
  float* persumsq = meanbuf + (size_t)Bn * Sn * CHn;         // B*S
  float* stdbuf   = persumsq + Bn * Sn;                      // B

  // FPS: 1 block / batch, 32 waves; 64KB + reduction slots of dynamic LDS.
  const size_t fps_lds = (size_t)(4 * Nn) * sizeof(float) + 2 * 32 * 4 + 16;
  fps_kernel<<<Bn, 1024, fps_lds, stream>>>(xyz, out, fps_idx);

  // kNN: 16 anchors per block, 256KB dynamic LDS panel (WGP has 320KB).
  const size_t knn_lds = (size_t)16 * Nn * sizeof(float) + 64 * sizeof(float);
  knn_kernel<<<Bn * (Sn / 16), 256, knn_lds, stream>>>(xyz, out, knn_idx);

  // Per-anchor mean + centered sum-of-squares, deterministic.
  stats_kernel<<<Bn * Sn, 128, 0, stream>>>(xyz, points, knn_idx, meanbuf, persumsq);

  // Per-batch std, deterministic tree.
  std_reduce_kernel<<<Bn, 1024, 0, stream>>>(persumsq, stdbuf);

  // Final 275MB output write.
  output_kernel<<<Bn * Sn, 256, 0, stream>>>(xyz, points, alpha, beta,
                                             fps_idx, knn_idx, meanbuf, stdbuf, out);
}